// GNNPolicy_9835475107962
// MI455X (gfx1250) — compile-verified
//
#include <hip/hip_runtime.h>
#include <hip/hip_bf16.h>

typedef __attribute__((ext_vector_type(2))) float v2f;
typedef __attribute__((ext_vector_type(8))) float v8f;

#define NODE_DIM 128
#define SW_STRIDE 136   // padded LDS row stride (dwords): rows k,k+2 differ by 16 banks

// ---------------------------------------------------------------------------
// Degree / normalization kernels
// ---------------------------------------------------------------------------
__global__ void gcn_deg_init(float* __restrict__ deg, int n) {
    int i = blockIdx.x * blockDim.x + threadIdx.x;
    if (i < n) deg[i] = 1.0f;                       // self-loop contribution
}

__global__ void gcn_deg_accum(const int* __restrict__ ei, float* __restrict__ deg, int E) {
    int e = blockIdx.x * blockDim.x + threadIdx.x;
    if (e < E) {
        int dst = ei[E + e];                        // edge_index[1][e]
        unsafeAtomicAdd(&deg[dst], 1.0f);
    }
}

__global__ void gcn_dinv(const float* __restrict__ deg, float* __restrict__ dinv, int n) {
    int i = blockIdx.x * blockDim.x + threadIdx.x;
    if (i < n) dinv[i] = rsqrtf(deg[i]);            // deg >= 1 always
}

// ---------------------------------------------------------------------------
// fp32 GEMM via V_WMMA_F32_16X16X4_F32: Y[M,128] = X[M,128] @ W[128,128].
// W staged once per block in LDS (padded stride, bank-conflict-free B reads).
// One wave computes a 16x128 strip: 8 accumulators, 8 WMMAs per A fragment.
// ---------------------------------------------------------------------------
__global__ void gcn_gemm_wmma(const float* __restrict__ X, const float* __restrict__ W,
                              float* __restrict__ Y, int M) {
    __shared__ float sW[NODE_DIM * SW_STRIDE];      // ~68 KB of the 320 KB WGP LDS

    const int tid  = threadIdx.x;
    const int wave = tid >> 5;
    const int lane = tid & 31;

    // Cooperative copy of W[128,128] into padded LDS (4096 float4 / 256 threads)
    for (int i = tid; i < NODE_DIM * (NODE_DIM / 4); i += 256) {
        int r = i >> 5;                             // row (128 rows)
        int c = (i & 31) << 2;                      // col in floats (32 float4 per row)
        *(float4*)(&sW[r * SW_STRIDE + c]) = *(const float4*)(W + r * NODE_DIM + c);
    }
    __syncthreads();

    const int mTile = blockIdx.x * 8 + wave;        // 8 waves per block, 1 strip each
    if (mTile >= (M >> 4)) return;                  // wave-uniform (EXEC all-1s at WMMA)

    const int row     = mTile * 16 + (lane & 15);   // A: M = lane%16
    const int kHalf   = (lane >> 4) * 2;            // lanes 16-31 hold K+2,K+3
    const int colBase = lane & 15;                  // B/C/D: N = lane%16

    v8f acc[8] = {};
    #pragma unroll 2
    for (int k = 0; k < NODE_DIM; k += 4) {
        v2f a = *(const v2f*)(X + (size_t)row * NODE_DIM + k + kHalf);
        const float* sWk = &sW[(k + kHalf) * SW_STRIDE];
        #pragma unroll
        for (int j = 0; j < 8; ++j) {
            v2f b;
            b.x = sWk[j * 16 + colBase];                 // row k+kHalf
            b.y = sWk[SW_STRIDE + j * 16 + colBase];     // row k+kHalf+1
            acc[j] = __builtin_amdgcn_wmma_f32_16x16x4_f32(false, a, false, b,
                                                           (short)0, acc[j], false, false);
        }
    }

    // C/D layout: VGPR v -> M = v (lanes 0-15) or v+8 (lanes 16-31)
    const int mBase = mTile * 16 + ((lane >> 4) << 3);
    #pragma unroll
    for (int j = 0; j < 8; ++j) {
        float* yj = Y + (size_t)mBase * NODE_DIM + j * 16 + colBase;
        #pragma unroll
        for (int v = 0; v < 8; ++v)
            yj[v * NODE_DIM] = acc[j][v];
    }
}

// ---------------------------------------------------------------------------
// out[i,:] = xw[i,:] * dinv[i]^2   (self-loop term; also initializes out)
// float4-vectorized: n4 = N*128/4 elements
// ---------------------------------------------------------------------------
__global__ void gcn_self_init(const float4* __restrict__ xw, const float* __restrict__ dinv,
                              float4* __restrict__ out, int n4) {
    int i = blockIdx.x * blockDim.x + threadIdx.x;
    if (i < n4) {
        float d = dinv[i >> 5];                     // 32 float4 per node row
        float s = d * d;
        float4 v = xw[i];
        out[i] = make_float4(v.x * s, v.y * s, v.z * s, v.w * s);
    }
}

// ---------------------------------------------------------------------------
// Edge scatter: out[dst,ch] += xw[src,ch] * dinv[src]*dinv[dst]
// One thread per (edge, channel): coalesced gather + native f32 atomics,
// all L2-resident (xw 25.6 MB + dinv 200 KB << 192 MB L2).
// ---------------------------------------------------------------------------
__global__ void gcn_edge_scatter(const int* __restrict__ ei, const float* __restrict__ xw,
                                 const float* __restrict__ dinv, float* __restrict__ out,
                                 int E) {
    int t = blockIdx.x * blockDim.x + threadIdx.x;  // E*128 = 64M < 2^31
    if (t < E * NODE_DIM) {
        int e   = t >> 7;
        int ch  = t & 127;
        int src = ei[e];
        int dst = ei[E + e];
        float v = xw[(size_t)src * NODE_DIM + ch] * dinv[src] * dinv[dst];
        unsafeAtomicAdd(&out[(size_t)dst * NODE_DIM + ch], v);
    }
}

// ---------------------------------------------------------------------------
// out = relu(out + b), float4-vectorized
// ---------------------------------------------------------------------------
__global__ void gcn_bias_relu(float4* __restrict__ out, const float4* __restrict__ b, int n4) {
    int i = blockIdx.x * blockDim.x + threadIdx.x;
    if (i < n4) {
        float4 v = out[i];
        float4 bb = b[i & 31];                      // 32 float4 per 128-ch row
        v.x = fmaxf(v.x + bb.x, 0.0f);
        v.y = fmaxf(v.y + bb.y, 0.0f);
        v.z = fmaxf(v.z + bb.z, 0.0f);
        v.w = fmaxf(v.w + bb.w, 0.0f);
        out[i] = v;
    }
}

// ---------------------------------------------------------------------------
extern "C" void kernel_launch(void* const* d_in, const int* in_sizes, int n_in,
                              void* d_out, int out_size, void* d_ws, size_t ws_size,
                              hipStream_t stream) {
    const float* x  = (const float*)d_in[0];
    const int*   ei = (const int*)  d_in[1];
    const float* W1 = (const float*)d_in[2];
    const float* b1 = (const float*)d_in[3];
    const float* W2 = (const float*)d_in[4];
    const float* b2 = (const float*)d_in[5];

    const int N  = in_sizes[0] / NODE_DIM;          // 50000
    const int E  = in_sizes[1] / 2;                 // 500000
    const int NH = N * NODE_DIM;                    // 6.4M
    const int N4 = NH / 4;

    float* deg  = (float*)d_ws;
    float* dinv = deg  + N;
    float* tmp1 = dinv + N;                         // xw buffer (N*128)
    float* tmp2 = tmp1 + (size_t)N * NODE_DIM;      // h  buffer (N*128)
    float* out  = (float*)d_out;

    const int T = 256;
    const int gemmBlocks = ((N >> 4) + 7) / 8;      // 1 strip per wave, 8 waves/block
    const int edgeBlocks = (E * NODE_DIM + T - 1) / T;
    const int v4Blocks   = (N4 + T - 1) / T;

    // --- degrees / normalization (shared by both layers) ---
    gcn_deg_init <<<(N + T - 1) / T, T, 0, stream>>>(deg, N);
    gcn_deg_accum<<<(E + T - 1) / T, T, 0, stream>>>(ei, deg, E);
    gcn_dinv     <<<(N + T - 1) / T, T, 0, stream>>>(deg, dinv, N);

    // --- layer 1: h = relu(norm-agg(x @ W1) + b1) ---
    gcn_gemm_wmma   <<<gemmBlocks, T, 0, stream>>>(x, W1, tmp1, N);
    gcn_self_init   <<<v4Blocks, T, 0, stream>>>((const float4*)tmp1, dinv, (float4*)tmp2, N4);
    gcn_edge_scatter<<<edgeBlocks, T, 0, stream>>>(ei, tmp1, dinv, tmp2, E);
    gcn_bias_relu   <<<v4Blocks, T, 0, stream>>>((float4*)tmp2, (const float4*)b1, N4);

    // --- layer 2: out = relu(norm-agg(h @ W2) + b2) ---
    gcn_gemm_wmma   <<<gemmBlocks, T, 0, stream>>>(tmp2, W2, tmp1, N);
    gcn_self_init   <<<v4Blocks, T, 0, stream>>>((const float4*)tmp1, dinv, (float4*)out, N4);
    gcn_edge_scatter<<<edgeBlocks, T, 0, stream>>>(ei, tmp1, dinv, out, E);
    gcn_bias_relu   <<<v4Blocks, T, 0, stream>>>((float4*)out, (const float4*)b2, N4);
}